// GATNet_9156870275401
// MI455X (gfx1250) — compile-verified
//
#include <hip/hip_runtime.h>
#include <hip/hip_bf16.h>

typedef __attribute__((ext_vector_type(2))) float v2f;
typedef __attribute__((ext_vector_type(8))) float v8f;

#define D_IN  512
#define D_HID 512
#define D_OUT 128
#define NEG_SLOPE 0.2f

// ---------------------------------------------------------------------------
// C[M,F] = A[M,K] @ B[K,F] in fp32 via V_WMMA_F32_16X16X4_F32.
// One wave (32 threads) per block computes a 16(M) x 64(N) tile.
// A frag (16x4 f32): lanes 0-15 -> M=lane, V0=K0,V1=K1 ; lanes 16-31 -> V0=K2,V1=K3
// B frag (4x16 f32): lanes 0-15 -> N=lane, V0=K0,V1=K1 ; lanes 16-31 -> V0=K2,V1=K3
// C/D (16x16 f32):   VGPR r, lanes 0-15 -> M=r,N=lane ; lanes 16-31 -> M=r+8
// ---------------------------------------------------------------------------
__global__ __launch_bounds__(32) void gat_gemm_f32_wmma(
    const float* __restrict__ A, const float* __restrict__ B,
    float* __restrict__ C, int M, int K, int F)
{
    const int lane  = threadIdx.x & 31;
    const int m0    = blockIdx.x * 16;
    const int n0    = blockIdx.y * 64;
    if (m0 >= M) return;                      // uniform per wave

    const int khalf = (lane >> 4) * 2;        // 0 or 2
    const int l15   = lane & 15;
    const float* Arow = A + (size_t)(m0 + l15) * K;

    v8f acc0 = {}; v8f acc1 = {}; v8f acc2 = {}; v8f acc3 = {};

    for (int k = 0; k < K; k += 4) {
        v2f a;
        a.x = Arow[k + khalf];
        a.y = Arow[k + khalf + 1];
        const float* Bk0 = B + (size_t)(k + khalf)     * F + n0 + l15;
        const float* Bk1 = B + (size_t)(k + khalf + 1) * F + n0 + l15;
        v2f b0, b1, b2, b3;
        b0.x = Bk0[0];  b0.y = Bk1[0];
        b1.x = Bk0[16]; b1.y = Bk1[16];
        b2.x = Bk0[32]; b2.y = Bk1[32];
        b3.x = Bk0[48]; b3.y = Bk1[48];
        acc0 = __builtin_amdgcn_wmma_f32_16x16x4_f32(false, a, false, b0, (short)0, acc0, false, false);
        acc1 = __builtin_amdgcn_wmma_f32_16x16x4_f32(false, a, false, b1, (short)0, acc1, false, false);
        acc2 = __builtin_amdgcn_wmma_f32_16x16x4_f32(false, a, false, b2, (short)0, acc2, false, false);
        acc3 = __builtin_amdgcn_wmma_f32_16x16x4_f32(false, a, false, b3, (short)0, acc3, false, false);
    }

    const int crow = m0 + (lane >> 4) * 8;
    float* Crow = C + (size_t)crow * F + n0 + l15;
    for (int r = 0; r < 8; ++r) {
        float* p = Crow + (size_t)r * F;
        p[0]  = acc0[r];
        p[16] = acc1[r];
        p[32] = acc2[r];
        p[48] = acc3[r];
    }
}

// ---------------------------------------------------------------------------
// alpha_s[n] = h[n,:] . a_src ; alpha_d[n] = h[n,:] . a_dst  (one wave / node)
// ---------------------------------------------------------------------------
__global__ __launch_bounds__(32) void gat_alpha(
    const float* __restrict__ h, const float* __restrict__ a_s,
    const float* __restrict__ a_d, float* __restrict__ as_,
    float* __restrict__ ad_, int F)
{
    const int node = blockIdx.x;
    const int lane = threadIdx.x;
    const float* row = h + (size_t)node * F;
    float s0 = 0.f, s1 = 0.f;
    for (int f = lane; f < F; f += 32) {
        float v = row[f];
        s0 += v * a_s[f];
        s1 += v * a_d[f];
    }
    for (int off = 16; off > 0; off >>= 1) {
        s0 += __shfl_down(s0, off, 32);
        s1 += __shfl_down(s1, off, 32);
    }
    if (lane == 0) { as_[node] = s0; ad_[node] = s1; }
}

// monotonic int encoding of float order: flip low 31 bits for negatives
__device__ __forceinline__ int f32_ordered(float f) {
    int i = __float_as_int(f);
    return i ^ ((i >> 31) & 0x7FFFFFFF);
}
__device__ __forceinline__ float f32_unordered(int i) {
    i ^= ((i >> 31) & 0x7FFFFFFF);
    return __int_as_float(i);
}

__global__ void gat_init_seg(int* __restrict__ mInt, float* __restrict__ denom, int N)
{
    int i = blockIdx.x * blockDim.x + threadIdx.x;
    if (i < N) { mInt[i] = 0x807FFFFF; denom[i] = 0.f; }   // ordered(-inf)
}

// e = leaky_relu(as[src] + ad[dst]); segment-max into mInt[dst]
__global__ void gat_edge_score(
    const int* __restrict__ src, const int* __restrict__ dst,
    const float* __restrict__ as_, const float* __restrict__ ad_,
    float* __restrict__ e, int* __restrict__ mInt, int E_real, int E_tot)
{
    int i = blockIdx.x * blockDim.x + threadIdx.x;
    if (i >= E_tot) return;
    int s, d;
    if (i < E_real) { s = src[i]; d = dst[i]; } else { s = d = i - E_real; }
    float v = as_[s] + ad_[d];
    v = v > 0.f ? v : NEG_SLOPE * v;
    e[i] = v;
    atomicMax(mInt + d, f32_ordered(v));
}

// e = exp(e - m[dst]); segment-sum into denom[dst]
__global__ void gat_edge_exp(
    const int* __restrict__ src, const int* __restrict__ dst,
    float* __restrict__ e, const int* __restrict__ mInt,
    float* __restrict__ denom, int E_real, int E_tot)
{
    int i = blockIdx.x * blockDim.x + threadIdx.x;
    if (i >= E_tot) return;
    int d = (i < E_real) ? dst[i] : (i - E_real);
    float m  = f32_unordered(mInt[d]);
    float ex = __expf(e[i] - m);
    e[i] = ex;
    atomicAdd(denom + d, ex);
}

// out[dst,:] += h[src,:] * (e / denom[dst])  — one block per edge
__global__ __launch_bounds__(128) void gat_edge_aggregate(
    const int* __restrict__ src, const int* __restrict__ dst,
    const float* __restrict__ e, const float* __restrict__ denom,
    const float* __restrict__ h, float* __restrict__ out,
    int E_real, int F)
{
    int edge = blockIdx.x;
    int s, d;
    if (edge < E_real) { s = src[edge]; d = dst[edge]; } else { s = d = edge - E_real; }
    float w = e[edge] / denom[d];
    const float* hr = h + (size_t)s * F;
    float* orow = out + (size_t)d * F;
    for (int f = threadIdx.x; f < F; f += 128)
        atomicAdd(orow + f, hr[f] * w);
}

__global__ void gat_bias_elu(float* __restrict__ x, const float* __restrict__ b, long long total, int Fmask)
{
    long long i = (long long)blockIdx.x * blockDim.x + threadIdx.x;
    if (i >= total) return;
    float v = x[i] + b[(int)i & Fmask];
    x[i] = v > 0.f ? v : (__expf(v) - 1.f);
}

__global__ void gat_bias_add(float* __restrict__ x, const float* __restrict__ b, long long total, int Fmask)
{
    long long i = (long long)blockIdx.x * blockDim.x + threadIdx.x;
    if (i >= total) return;
    x[i] += b[(int)i & Fmask];
}

// ---------------------------------------------------------------------------
extern "C" void kernel_launch(void* const* d_in, const int* in_sizes, int n_in,
                              void* d_out, int out_size, void* d_ws, size_t ws_size,
                              hipStream_t stream) {
    const float* feature = (const float*)d_in[0];
    const int*   edges   = (const int*)d_in[1];
    const float* W1      = (const float*)d_in[2];
    const float* a_src1  = (const float*)d_in[3];
    const float* a_dst1  = (const float*)d_in[4];
    const float* b1      = (const float*)d_in[5];
    const float* W2      = (const float*)d_in[6];
    const float* a_src2  = (const float*)d_in[7];
    const float* a_dst2  = (const float*)d_in[8];
    const float* b2      = (const float*)d_in[9];
    float* out = (float*)d_out;

    const int N      = in_sizes[0] / D_IN;     // 10000
    const int E_real = in_sizes[1] / 2;        // 160000
    const int E_tot  = E_real + N;             // with self-loops
    const int* src = edges;
    const int* dst = edges + E_real;

    // workspace layout (h2 aliases h1: h1 is dead once layer-1 aggregation ends)
    float* h1    = (float*)d_ws;                    // N*D_HID
    float* agg1  = h1   + (size_t)N * D_HID;        // N*D_HID (becomes layer-2 input)
    float* as_   = agg1 + (size_t)N * D_HID;        // N
    float* ad_   = as_  + N;                        // N
    float* denom = ad_  + N;                        // N
    int*   mInt  = (int*)(denom + N);               // N
    float* ebuf  = (float*)(mInt + N);              // E_tot
    float* h2    = h1;                              // N*D_OUT (alias)

    const int T = 256;
    dim3 blk32(32), blk128(128), blkT(T);
    dim3 gN((unsigned)N), gE((unsigned)E_tot);
    dim3 gEdgeT((E_tot + T - 1) / T), gNT((N + T - 1) / T);

    // ---------------- Layer 1 ----------------
    gat_gemm_f32_wmma<<<dim3(N / 16, D_HID / 64), blk32, 0, stream>>>(feature, W1, h1, N, D_IN, D_HID);
    gat_alpha<<<gN, blk32, 0, stream>>>(h1, a_src1, a_dst1, as_, ad_, D_HID);
    gat_init_seg<<<gNT, blkT, 0, stream>>>(mInt, denom, N);
    hipMemsetAsync(agg1, 0, (size_t)N * D_HID * sizeof(float), stream);
    gat_edge_score<<<gEdgeT, blkT, 0, stream>>>(src, dst, as_, ad_, ebuf, mInt, E_real, E_tot);
    gat_edge_exp<<<gEdgeT, blkT, 0, stream>>>(src, dst, ebuf, mInt, denom, E_real, E_tot);
    gat_edge_aggregate<<<gE, blk128, 0, stream>>>(src, dst, ebuf, denom, h1, agg1, E_real, D_HID);
    {
        long long total = (long long)N * D_HID;
        gat_bias_elu<<<dim3((unsigned)((total + T - 1) / T)), blkT, 0, stream>>>(agg1, b1, total, D_HID - 1);
    }

    // ---------------- Layer 2 ----------------
    gat_gemm_f32_wmma<<<dim3(N / 16, D_OUT / 64), blk32, 0, stream>>>(agg1, W2, h2, N, D_HID, D_OUT);
    gat_alpha<<<gN, blk32, 0, stream>>>(h2, a_src2, a_dst2, as_, ad_, D_OUT);
    gat_init_seg<<<gNT, blkT, 0, stream>>>(mInt, denom, N);
    hipMemsetAsync(out, 0, (size_t)N * D_OUT * sizeof(float), stream);
    gat_edge_score<<<gEdgeT, blkT, 0, stream>>>(src, dst, as_, ad_, ebuf, mInt, E_real, E_tot);
    gat_edge_exp<<<gEdgeT, blkT, 0, stream>>>(src, dst, ebuf, mInt, denom, E_real, E_tot);
    gat_edge_aggregate<<<gE, blk128, 0, stream>>>(src, dst, ebuf, denom, h2, out, E_real, D_OUT);
    {
        long long total = (long long)N * D_OUT;
        gat_bias_add<<<dim3((unsigned)((total + T - 1) / T)), blkT, 0, stream>>>(out, b2, total, D_OUT - 1);
    }
}